// HistogramLayer_41051297415449
// MI455X (gfx1250) — compile-verified
//
#include <hip/hip_runtime.h>
#include <hip/hip_bf16.h>

#define NUM_BINS 16
#define NUM_EDGES 17
#define DFEAT 512

typedef unsigned int  u32;
typedef unsigned long long u64;
typedef u32 v4u __attribute__((ext_vector_type(4)));
typedef int v8i __attribute__((ext_vector_type(8)));
typedef int v4i __attribute__((ext_vector_type(4)));

// ---------------------------------------------------------------------------
// Tensor Data Mover: DMA a 2D row-major f32 tile [dim1 rows x dim0 cols]
// (row pitch = stride0 elements) from global memory into LDS at lds_off.
// D# layout per CDNA5 ISA ch.8 (group0 128b, group1 256b; groups 2/3 unused
// for 2D). data_size=2 -> 4-byte elements; no padding, no multicast,
// no atomic-barrier, no iteration, no gather.
// This toolchain exposes the 6-arg builtin (extra int32x8 group, zeroed).
// ---------------------------------------------------------------------------
__device__ __forceinline__ void tdm_load_tile_2d_f32(u32 lds_off,
                                                     const void* gptr,
                                                     u32 dim0, u32 dim1,
                                                     u32 stride0) {
    const u64 ga = (u64)gptr;
    v4u g0;
    g0.x = 1u;                                          // count=1, is_restore=0
    g0.y = lds_off;                                     // lds_addr[31:0]
    g0.z = (u32)(ga & 0xFFFFFFFFu);                     // global_addr[31:0]
    g0.w = (u32)((ga >> 32) & 0x01FFFFFFu) | (2u << 30); // global_addr[56:32] | type=2

    v8i g1;
    g1[0] = (int)(2u << 16);                                  // data_size=4B; wg_mask=0
    g1[1] = (int)((dim0 & 0xFFFFu) << 16);                    // tensor_dim0[15:0] @ [63:48]
    g1[2] = (int)(((dim0 >> 16) & 0xFFFFu) |                  // tensor_dim0[31:16]
                  ((dim1 & 0xFFFFu) << 16));                  // tensor_dim1[15:0]
    g1[3] = (int)(((dim1 >> 16) & 0xFFFFu) |                  // tensor_dim1[31:16]
                  ((dim0 & 0xFFFFu) << 16));                  // tile_dim0 @ [127:112]
    g1[4] = (int)(dim1 & 0xFFFFu);                            // tile_dim1; tile_dim2=0
    g1[5] = (int)stride0;                                     // tensor_dim0_stride[31:0]
    g1[6] = 0;                                                // stride0[47:32]; stride1 lo
    g1[7] = 0;

    const v4i z4 = {0, 0, 0, 0};
    const v8i z8 = {0, 0, 0, 0, 0, 0, 0, 0};
    __builtin_amdgcn_tensor_load_to_lds(g0, g1, z4, z4, z8, 0);
}

// ---------------------------------------------------------------------------
// One wave (32 lanes) per row. Lane l covers features l, l+32, ... l+480:
// LDS row pitch is 512 floats (512 % 64banks == 0), so every table access is
// bank-conflict free across the wave, including the divergent search probes.
// ---------------------------------------------------------------------------
__global__ __launch_bounds__(256) void HistogramLayer_41051297415449_kernel(
    const float* __restrict__ inputs,     // [B, 512]
    const float* __restrict__ frequencies,// [16, 512]
    const float* __restrict__ edges,      // [17, 512]
    float* __restrict__ out,              // [B]
    int B) {
    __shared__ float s_edges[NUM_EDGES * DFEAT];  // 34816 B
    __shared__ float s_probs[NUM_BINS * DFEAT];   // 32768 B

    // --- async tile DMA (wave 0 only), then block-wide barrier -------------
    if (threadIdx.x < 32u) {
        tdm_load_tile_2d_f32((u32)(u64)(void*)s_edges, edges, DFEAT, NUM_EDGES, DFEAT);
        tdm_load_tile_2d_f32((u32)(u64)(void*)s_probs, frequencies, DFEAT, NUM_BINS, DFEAT);
        __builtin_amdgcn_s_wait_tensorcnt(0);
    }
    __syncthreads();

    // --- normalize frequencies -> probabilities in place in LDS ------------
    for (int col = threadIdx.x; col < DFEAT; col += blockDim.x) {
        float s = 0.0f;
#pragma unroll
        for (int b = 0; b < NUM_BINS; ++b) s += s_probs[b * DFEAT + col];
        const float inv = 1.0f / s;
#pragma unroll
        for (int b = 0; b < NUM_BINS; ++b) s_probs[b * DFEAT + col] *= inv;
    }
    __syncthreads();

    const int lane        = threadIdx.x & 31;
    const int waveInBlk   = threadIdx.x >> 5;
    const int wavesPerBlk = blockDim.x >> 5;
    const int waveGlobal  = blockIdx.x * wavesPerBlk + waveInBlk;
    const int wavesTotal  = gridDim.x * wavesPerBlk;

    for (int row = waveGlobal; row < B; row += wavesTotal) {
        const float* rp = inputs + (size_t)row * DFEAT;
        float p = 1.0f;
#pragma unroll 4
        for (int k = 0; k < DFEAT / 32; ++k) {
            const int f = lane + 32 * k;
            const float x = __builtin_nontemporal_load(rp + f);  // one-pass stream: TH=NT
            // rank = #{edges[e][f] <= x}: branchless binary search over the
            // first 16 edges (5 probes) + probe of edge 16.
            int c = 0;
            c += (s_edges[(c + 7) * DFEAT + f] <= x) ? 8 : 0;
            c += (s_edges[(c + 3) * DFEAT + f] <= x) ? 4 : 0;
            c += (s_edges[(c + 1) * DFEAT + f] <= x) ? 2 : 0;
            c += (s_edges[(c + 0) * DFEAT + f] <= x) ? 1 : 0;
            c += (s_edges[(c + 0) * DFEAT + f] <= x) ? 1 : 0;  // window of 1
            c += (s_edges[16 * DFEAT + f] <= x) ? 1 : 0;       // 17th edge
            int idx = c - 1;                                   // searchsorted_right - 1
            idx = idx < 0 ? 0 : (idx > NUM_BINS - 1 ? NUM_BINS - 1 : idx);
            p *= s_probs[idx * DFEAT + f];
        }
        // wave32 product reduction (lane permute, no LDS)
#pragma unroll
        for (int off = 16; off > 0; off >>= 1) p *= __shfl_xor(p, off, 32);
        if (lane == 0) out[row] = p;
    }
}

extern "C" void kernel_launch(void* const* d_in, const int* in_sizes, int n_in,
                              void* d_out, int out_size, void* d_ws, size_t ws_size,
                              hipStream_t stream) {
    const float* inputs      = (const float*)d_in[0];  // [B, 512]
    const float* frequencies = (const float*)d_in[1];  // [16, 512]
    const float* edges       = (const float*)d_in[2];  // [17, 512]
    float* out = (float*)d_out;

    const int B = in_sizes[0] / DFEAT;

    // 512 blocks x 8 wave32s: LDS-limited to 4 blocks/WGP (67.5 KB each),
    // grid-stride over rows (16 rows per wave at B=65536).
    const int blocks = 512;
    HistogramLayer_41051297415449_kernel<<<blocks, 256, 0, stream>>>(
        inputs, frequencies, edges, out, B);
}